// FeatureTransformer_51685636440709
// MI455X (gfx1250) — compile-verified
//
#include <hip/hip_runtime.h>
#include <math.h>

typedef float v2f __attribute__((ext_vector_type(2)));
typedef float v8f __attribute__((ext_vector_type(8)));

#define ACT_NONE 0
#define ACT_RELU 1
#define ACT_GELU 2
#define ACT_SIG  3
#define ACT_TANH 4

struct Taps9 { int dy[9]; int dx[9]; };

template<int ACT>
__device__ __forceinline__ float act_f(float x){
  if constexpr (ACT==ACT_RELU) return fmaxf(x,0.0f);
  else if constexpr (ACT==ACT_GELU) return 0.5f*x*(1.0f+erff(x*0.70710678118654752440f));
  else if constexpr (ACT==ACT_SIG)  return 1.0f/(1.0f+expf(-x));
  else if constexpr (ACT==ACT_TANH) return tanhf(x);
  else return x;
}

// Full-precision f32 matrix core op: D(16x16) = A(16x4) * B(4x16) + C
__device__ __forceinline__ v8f wmma_k4(v2f a, v2f b, v8f c){
  return __builtin_amdgcn_wmma_f32_16x16x4_f32(false, a, false, b, (short)0, c, false, false);
}

// ---------------------------------------------------------------------------
// GEMM: C[b][M,N] = act( A[b][M,K] @ Bw[b][N,K]^T + bias[N] )
// block = 256 threads = 8 waves (4 in M x 2 in N); wave tile = 32M x 32N
// (4 accumulators -> 4 WMMAs per 4 loads per k-step; A/B fragments each
//  reused twice). Requires M%128==0, N%64==0, K%4==0 (always true here).
// A-fragment (16x4 f32): lanes 0-15 hold K={k,k+1} of row m0+lane,
//                        lanes 16-31 hold K={k+2,k+3}  -> one 64-bit load/lane.
// B-fragment (4x16): same K-split across lane halves, N = lane&15 -> 64-bit load.
// D layout: VGPR r -> row m0 + r + (lane<16?0:8), col n0 + (lane&15).
// ---------------------------------------------------------------------------
template<int ACT>
__global__ __launch_bounds__(256) void k_gemm(
    const float* __restrict__ A, long sA,
    const float* __restrict__ Bw, long sB,
    const float* __restrict__ bias,
    float* __restrict__ C, long sC,
    int M, int N, int K)
{
  (void)M;
  const int lane  = threadIdx.x & 31;
  const int wave  = threadIdx.x >> 5;
  const int l15   = lane & 15;
  const int half2 = (lane >> 4) << 1;            // 0 or 2
  const int m0 = blockIdx.x * 128 + (wave & 3) * 32;
  const int n0 = blockIdx.y * 64 + (wave >> 2) * 32;
  const float* A0 = A  + (long)blockIdx.z * sA + (long)(m0 + l15) * K + half2;
  const float* A1 = A0 + (long)16 * K;
  const float* B0 = Bw + (long)blockIdx.z * sB + (long)(n0 + l15) * K + half2;
  const float* B1 = B0 + (long)16 * K;
  __builtin_prefetch(A0, 0, 3);                  // global_prefetch_b8
  __builtin_prefetch(B0, 0, 3);
  v8f acc00 = {}; v8f acc01 = {}; v8f acc10 = {}; v8f acc11 = {};
#pragma unroll 4
  for (int k = 0; k < K; k += 4) {
    v2f a0 = *(const v2f*)(A0 + k);
    v2f a1 = *(const v2f*)(A1 + k);
    v2f b0 = *(const v2f*)(B0 + k);
    v2f b1 = *(const v2f*)(B1 + k);
    acc00 = wmma_k4(a0, b0, acc00);
    acc01 = wmma_k4(a0, b1, acc01);
    acc10 = wmma_k4(a1, b0, acc10);
    acc11 = wmma_k4(a1, b1, acc11);
  }
  float* Cb = C + (long)blockIdx.z * sC;
  const int rbase = (lane >> 4) * 8;
  const bool hb = (bias != nullptr);
  const int n1 = n0 + l15;
  const float bv0 = hb ? bias[n1] : 0.0f;
  const float bv1 = hb ? bias[n1 + 16] : 0.0f;
#pragma unroll
  for (int r = 0; r < 8; ++r) {
    const int m = m0 + rbase + r;
    float* row0 = Cb + (long)m * N;
    float* row1 = row0 + (long)16 * N;
    row0[n1]      = act_f<ACT>(acc00[r] + bv0);
    row0[n1 + 16] = act_f<ACT>(acc01[r] + bv1);
    row1[n1]      = act_f<ACT>(acc10[r] + bv0);
    row1[n1 + 16] = act_f<ACT>(acc11[r] + bv1);
  }
}

// ---------------------------------------------------------------------------
// Direct conv (NHWC, im2col-on-the-fly) via f32 WMMA.
// X: [b][H*W, Cin]; WT: [NT][Cout][Cin] (pre-transposed); Y: [b][H*W, Cout].
// Out-of-bounds taps read a zeroed page through a per-lane pointer select
// (v_cndmask on the address) -> no EXEC manipulation, no branches in the
// hot loop, loads always execute and WMMA sees EXEC == all-ones.
// ---------------------------------------------------------------------------
template<int NT, int ACT>
__global__ __launch_bounds__(256) void k_conv(
    const float* __restrict__ X, long sX, int Cin,
    const float* __restrict__ WT,
    const float* __restrict__ bias,
    const float* __restrict__ zpage,
    float* __restrict__ Y, long sY, int Cout,
    Taps9 taps, int H, int W)
{
  const int lane  = threadIdx.x & 31;
  const int wave  = threadIdx.x >> 5;
  const int l15   = lane & 15;
  const int half2 = (lane >> 4) << 1;
  const int m0 = blockIdx.x * 128 + (wave & 3) * 32;
  const int n0 = blockIdx.y * 64 + (wave >> 2) * 32;
  const int mA = m0 + l15, mB = mA + 16;
  const int pyA = mA / W, pxA = mA % W;
  const int pyB = mB / W, pxB = mB % W;
  const float* Xb = X + (long)blockIdx.z * sX;
  const float* zp = zpage + half2;
  v8f acc00 = {}; v8f acc01 = {}; v8f acc10 = {}; v8f acc11 = {};
#pragma unroll
  for (int t = 0; t < NT; ++t) {
    const int yA = pyA + taps.dy[t], xA = pxA + taps.dx[t];
    const int yB = pyB + taps.dy[t], xB = pxB + taps.dx[t];
    const bool okA = ((unsigned)yA < (unsigned)H) && ((unsigned)xA < (unsigned)W);
    const bool okB = ((unsigned)yB < (unsigned)H) && ((unsigned)xB < (unsigned)W);
    const float* xr0 = okA ? (Xb + (long)(yA * W + xA) * Cin + half2) : zp;
    const float* xr1 = okB ? (Xb + (long)(yB * W + xB) * Cin + half2) : zp;
    const float* w0 = WT + ((long)t * Cout + n0 + l15) * Cin + half2;
    const float* w1 = w0 + (long)16 * Cin;
#pragma unroll 4
    for (int k = 0; k < Cin; k += 4) {
      v2f a0 = *(const v2f*)(xr0 + k);
      v2f a1 = *(const v2f*)(xr1 + k);
      v2f b0 = *(const v2f*)(w0 + k);
      v2f b1 = *(const v2f*)(w1 + k);
      acc00 = wmma_k4(a0, b0, acc00);
      acc01 = wmma_k4(a0, b1, acc01);
      acc10 = wmma_k4(a1, b0, acc10);
      acc11 = wmma_k4(a1, b1, acc11);
    }
  }
  float* Yb = Y + (long)blockIdx.z * sY;
  const int rbase = (lane >> 4) * 8;
  const int n1 = n0 + l15;
  const float bv0 = bias[n1];
  const float bv1 = bias[n1 + 16];
#pragma unroll
  for (int r = 0; r < 8; ++r) {
    const int m = m0 + rbase + r;
    float* row0 = Yb + (long)m * Cout;
    float* row1 = row0 + (long)16 * Cout;
    row0[n1]      = act_f<ACT>(acc00[r] + bv0);
    row0[n1 + 16] = act_f<ACT>(acc01[r] + bv1);
    row1[n1]      = act_f<ACT>(acc10[r] + bv0);
    row1[n1 + 16] = act_f<ACT>(acc11[r] + bv1);
  }
}

// ------------------------------ utility kernels ----------------------------

__global__ void k_zero(float* p, int n){
  int i = blockIdx.x * blockDim.x + threadIdx.x;
  if (i < n) p[i] = 0.0f;
}

// NCHW -> NHWC, duplicated into att and value
__global__ void k_in_tr(const float* __restrict__ in, float* __restrict__ att,
                        float* __restrict__ val, int C, int L, long n){
  long idx = (long)blockIdx.x * blockDim.x + threadIdx.x;
  if (idx >= n) return;
  const long LC = (long)L * C;
  int b = (int)(idx / LC); long rem = idx - (long)b * LC;
  int l = (int)(rem / C), c = (int)(rem % C);
  float v = in[(long)b * LC + (long)c * L + l];
  att[idx] = v; val[idx] = v;
}

// [b][R,C] -> [b][C,R]
__global__ void k_transpose(const float* __restrict__ X, float* __restrict__ Y,
                            int R, int C, long n){
  long idx = (long)blockIdx.x * blockDim.x + threadIdx.x;
  if (idx >= n) return;
  const long RC = (long)R * C;
  int b = (int)(idx / RC); long rem = idx - (long)b * RC;
  int r = (int)(rem / C), c = (int)(rem % C);
  Y[(long)b * RC + (long)c * R + r] = X[idx];
}

// OIHW [Co][Ci][T] -> [T][Co][Ci]
__global__ void k_wtrans(const float* __restrict__ Wsrc, float* __restrict__ WT,
                         int Co, int Ci, int T, long n){
  long idx = (long)blockIdx.x * blockDim.x + threadIdx.x;
  if (idx >= n) return;
  int co = (int)(idx / ((long)Ci * T)); long rem = idx - (long)co * Ci * T;
  int ci = (int)(rem / T), t = (int)(rem % T);
  WT[((long)t * Co + co) * Ci + ci] = Wsrc[idx];
}

// per-row (C=128): norm_fac = sqrt(sumsq), rsum = sum  (wave per row)
__global__ void k_rowstat(const float* __restrict__ X, float* __restrict__ nf,
                          float* __restrict__ rsum, int rows){
  int row = blockIdx.x * 8 + (threadIdx.x >> 5);
  int lane = threadIdx.x & 31;
  if (row >= rows) return;
  const float* x = X + (long)row * 128;
  float s = 0.0f, q = 0.0f;
#pragma unroll
  for (int i = 0; i < 4; ++i){ float v = x[lane * 4 + i]; s += v; q += v * v; }
  for (int o = 16; o > 0; o >>= 1){ s += __shfl_xor(s, o); q += __shfl_xor(q, o); }
  if (lane == 0){ nf[row] = sqrtf(q); rsum[row] = s; }
}

// one block per batch: deterministic tree-reduce L row-sums.
// mode 1: scale = 3*pi*sigmoid(mean)+1; mode 0: plain sum.
__global__ void k_batchred(const float* __restrict__ rows, float* __restrict__ out,
                           int L, int mode, float invLC){
  int b = blockIdx.x;
  float s = 0.0f;
  for (int i = threadIdx.x; i < L; i += blockDim.x) s += rows[(long)b * L + i];
  __shared__ float sm[8];
  int lane = threadIdx.x & 31, wave = threadIdx.x >> 5;
  for (int o = 16; o > 0; o >>= 1) s += __shfl_xor(s, o);
  if (lane == 0) sm[wave] = s;
  __syncthreads();
  if (threadIdx.x == 0){
    float t = 0.0f;
    for (int i = 0; i < 8; ++i) t += sm[i];
    if (mode == 1)
      t = 3.0f * 3.14159265358979323846f * (1.0f / (1.0f + expf(-t * invLC))) + 1.0f;
    out[b] = t;
  }
}

__global__ void k_an(const float* __restrict__ ap, const float* __restrict__ nf,
                     float* __restrict__ an, long n){
  long idx = (long)blockIdx.x * blockDim.x + threadIdx.x;
  if (idx >= n) return;
  an[idx] = ap[idx] / nf[idx >> 7];   // C=128
}

// A = exp(scale[b]*A); track global max (positive floats -> uint atomicMax).
__global__ void k_expmax(float* __restrict__ A, const float* __restrict__ scale,
                         unsigned* __restrict__ gmax, long n, long LL){
  long idx = (long)blockIdx.x * blockDim.x + threadIdx.x;
  float v = 0.0f;
  if (idx < n){
    int b = (int)(idx / LL);
    v = expf(scale[b] * A[idx]);
    A[idx] = v;
  }
  for (int o = 16; o > 0; o >>= 1) v = fmaxf(v, __shfl_xor(v, o));
  __shared__ float sm[8];
  int lane = threadIdx.x & 31, wave = threadIdx.x >> 5;
  if (lane == 0) sm[wave] = v;
  __syncthreads();
  if (wave == 0){
    float w = (lane < 8) ? sm[lane] : 0.0f;
    for (int o = 4; o > 0; o >>= 1) w = fmaxf(w, __shfl_xor(w, o));
    if (lane == 0) atomicMax(gmax, __float_as_uint(w));
  }
}

// wave per row, generic length row-sum
__global__ void k_rowsum(const float* __restrict__ X, float* __restrict__ out,
                         int len, int rows){
  int row = blockIdx.x * 8 + (threadIdx.x >> 5);
  int lane = threadIdx.x & 31;
  if (row >= rows) return;
  const float* x = X + (long)row * len;
  float s = 0.0f;
  for (int i = lane; i < len; i += 32) s += x[i];
  for (int o = 16; o > 0; o >>= 1) s += __shfl_xor(s, o);
  if (lane == 0) out[row] = s;
}

__global__ void k_dg(const float* __restrict__ rs, const unsigned* __restrict__ gmax,
                     float* __restrict__ Dg, int n){
  int i = blockIdx.x * blockDim.x + threadIdx.x;
  if (i >= n) return;
  float gm = __uint_as_float(*gmax);
  Dg[i] = 1.0f / (sqrtf(rs[i] / gm) + 1e-6f);
}

__global__ void k_anorm(float* __restrict__ A, const float* __restrict__ Dg,
                        const unsigned* __restrict__ gmax, long n, int L, long LL){
  long idx = (long)blockIdx.x * blockDim.x + threadIdx.x;
  if (idx >= n) return;
  int b = (int)(idx / LL); long rem = idx - (long)b * LL;
  int l = (int)(rem / L), m = (int)(rem % L);
  float gm = __uint_as_float(*gmax);
  A[idx] = A[idx] / gm * Dg[b * L + l] * Dg[b * L + m];
}

// LayerNorm over C=128, wave per row
__global__ void k_ln(const float* __restrict__ X, const float* __restrict__ g,
                     const float* __restrict__ bb, float* __restrict__ Y, int rows){
  int row = blockIdx.x * 8 + (threadIdx.x >> 5);
  int lane = threadIdx.x & 31;
  if (row >= rows) return;
  const float* x = X + (long)row * 128;
  float v[4]; float s = 0.0f;
#pragma unroll
  for (int i = 0; i < 4; ++i){ v[i] = x[lane * 4 + i]; s += v[i]; }
  for (int o = 16; o > 0; o >>= 1) s += __shfl_xor(s, o);
  float mu = s * (1.0f / 128.0f);
  float q = 0.0f;
#pragma unroll
  for (int i = 0; i < 4; ++i){ float d = v[i] - mu; q += d * d; }
  for (int o = 16; o > 0; o >>= 1) q += __shfl_xor(q, o);
  float inv = rsqrtf(q * (1.0f / 128.0f) + 1e-5f);
  float* y = Y + (long)row * 128;
#pragma unroll
  for (int i = 0; i < 4; ++i){ int c = lane * 4 + i; y[c] = (v[i] - mu) * inv * g[c] + bb[c]; }
}

__global__ void k_cat2(const float* __restrict__ a, const float* __restrict__ b,
                       float* __restrict__ o, long rows, int C){
  long idx = (long)blockIdx.x * blockDim.x + threadIdx.x;
  long n = rows * 2 * C;
  if (idx >= n) return;
  long row = idx / (2 * C); int c = (int)(idx - row * 2 * C);
  o[idx] = (c < C) ? a[row * C + c] : b[row * C + (c - C)];
}

__global__ void k_catrh(const float* __restrict__ r, const float* __restrict__ h,
                        const float* __restrict__ x, float* __restrict__ o,
                        long rows, int C){
  long idx = (long)blockIdx.x * blockDim.x + threadIdx.x;
  long n = rows * 2 * C;
  if (idx >= n) return;
  long row = idx / (2 * C); int c = (int)(idx - row * 2 * C);
  o[idx] = (c < C) ? r[row * C + c] * h[row * C + c] : x[row * C + (c - C)];
}

__global__ void k_gru(float* __restrict__ h, const float* __restrict__ z,
                      const float* __restrict__ q, long n){
  long idx = (long)blockIdx.x * blockDim.x + threadIdx.x;
  if (idx >= n) return;
  float zv = z[idx];
  h[idx] = (1.0f - zv) * h[idx] + zv * q[idx];
}

// NHWC -> NCHW with |k| * x / (mean + sigma^2)
__global__ void k_final(const float* __restrict__ X, const float* __restrict__ dsum,
                        const float* __restrict__ sigma, const float* __restrict__ kk,
                        float* __restrict__ out, int C, int L, long n, float invLC){
  long idx = (long)blockIdx.x * blockDim.x + threadIdx.x;
  if (idx >= n) return;
  long CL = (long)C * L;
  int b = (int)(idx / CL); long rem = idx - (long)b * CL;
  int c = (int)(rem / L), l = (int)(rem % L);
  float dn = dsum[b] * invLC + sigma[0] * sigma[0];
  out[idx] = fabsf(kk[0]) * X[((long)b * L + l) * C + c] / dn;
}

// ---------------------------------------------------------------------------

static void run_sep_gru(hipStream_t s, float* h, const float* x,
                        float* const* wt, const float* const* bias,
                        const float* zpage,
                        float* catA, float* catB, float* z, float* r, float* q,
                        const Taps9& tH, const Taps9& tV)
{
  const int Bn = 2, C = 128, HH = 48, WW = 48, L = HH * WW;
  const long LC = (long)L * C, L2C = 2 * LC;
  const long nLC = Bn * LC, nL2C = Bn * L2C;
  dim3 gc(18, 2, Bn);
  dim3 e2((unsigned)((nL2C + 255) / 256)), e1((unsigned)((nLC + 255) / 256));
  k_cat2<<<e2, 256, 0, s>>>(h, x, catA, (long)Bn * L, C);
  k_conv<5, ACT_SIG ><<<gc, 256, 0, s>>>(catA, L2C, 256, wt[0], bias[0], zpage, z, LC, C, tH, HH, WW);
  k_conv<5, ACT_SIG ><<<gc, 256, 0, s>>>(catA, L2C, 256, wt[1], bias[1], zpage, r, LC, C, tH, HH, WW);
  k_catrh<<<e2, 256, 0, s>>>(r, h, x, catB, (long)Bn * L, C);
  k_conv<5, ACT_TANH><<<gc, 256, 0, s>>>(catB, L2C, 256, wt[2], bias[2], zpage, q, LC, C, tH, HH, WW);
  k_gru<<<e1, 256, 0, s>>>(h, z, q, nLC);
  k_cat2<<<e2, 256, 0, s>>>(h, x, catA, (long)Bn * L, C);
  k_conv<5, ACT_SIG ><<<gc, 256, 0, s>>>(catA, L2C, 256, wt[3], bias[3], zpage, z, LC, C, tV, HH, WW);
  k_conv<5, ACT_SIG ><<<gc, 256, 0, s>>>(catA, L2C, 256, wt[4], bias[4], zpage, r, LC, C, tV, HH, WW);
  k_catrh<<<e2, 256, 0, s>>>(r, h, x, catB, (long)Bn * L, C);
  k_conv<5, ACT_TANH><<<gc, 256, 0, s>>>(catB, L2C, 256, wt[5], bias[5], zpage, q, LC, C, tV, HH, WW);
  k_gru<<<e1, 256, 0, s>>>(h, z, q, nLC);
}

extern "C" void kernel_launch(void* const* d_in, const int* in_sizes, int n_in,
                              void* d_out, int out_size, void* d_ws, size_t ws_size,
                              hipStream_t stream) {
  (void)in_sizes; (void)out_size; (void)ws_size;
  if (n_in < 51) return;
  auto F = [&](int i){ return (const float*)d_in[i]; };

  const int Bn = 2, C = 128, HH = 48, WW = 48, L = HH * WW;      // L = 2304
  const long LC   = (long)L * C;        // per-batch [L,C]
  const long L2C  = 2 * LC;
  const long L512 = (long)L * 512;
  const long LL   = (long)L * L;
  auto ew = [](long n){ return dim3((unsigned)((n + 255) / 256)); };

  // ---- scratch bump allocator over d_ws ----
  char* p = (char*)d_ws;
  auto alloc = [&](long nfloats)->float*{
    float* r = (float*)p; p += ((nfloats * 4 + 255) / 256) * 256; return r;
  };
  float* sc       = alloc(16);            // [0..1]=scale [2]=gmax(bits) [3..4]=denom
  float* zpage    = alloc(512);           // zero page for conv OOB taps
  float* normfac  = alloc((long)Bn * L);
  float* rowtmp   = alloc((long)Bn * L);
  float* Dg       = alloc((long)Bn * L);
  float* att  = alloc(Bn * LC);
  float* val  = alloc(Bn * LC);
  float* vp   = alloc(Bn * LC);
  float* vpT  = alloc(Bn * LC);
  float* t1   = alloc(Bn * LC);
  float* ap   = alloc(Bn * LC);
  float* an   = alloc(Bn * LC);
  float* Abig = alloc(Bn * LL);
  float* bufA = alloc(Bn * LC);
  float* bufB = alloc(Bn * LC);
  float* bufC = alloc(Bn * LC);           // msg2
  float* bufD = alloc(Bn * LC);
  float* bufE = alloc(Bn * LC);
  float* cat0 = alloc(Bn * L2C);
  float* cat1 = alloc(Bn * L2C);
  float* h1   = alloc(Bn * L512);
  float* h2   = alloc(Bn * L512);
  float* zb   = alloc(Bn * LC);
  float* rb   = alloc(Bn * LC);
  float* qb   = alloc(Bn * LC);
  float* wt_c1 = alloc((long)1 * 128 * 256);
  float* wt_c2 = alloc((long)9 * 128 * 128);
  float* wt_c3 = alloc((long)9 * 128 * 128);
  float* wt_gru[12];
  for (int i = 0; i < 12; ++i) wt_gru[i] = alloc((long)5 * 128 * 256);

  k_zero<<<2, 256, 0, stream>>>(zpage, 512);

  // ---- taps ----
  Taps9 taps1{}; Taps9 tapsH{}; Taps9 tapsV{}; Taps9 taps3{};
  for (int t = 0; t < 5; ++t){ tapsH.dy[t] = 0; tapsH.dx[t] = t - 2;
                               tapsV.dy[t] = t - 2; tapsV.dx[t] = 0; }
  for (int t = 0; t < 9; ++t){ taps3.dy[t] = t / 3 - 1; taps3.dx[t] = t % 3 - 1; }

  // ---- pre-transpose conv weights: OIHW -> [tap][Cout][Cin] (once, reused by both layers)
  auto wtr = [&](const float* w, float* dst, int Co, int Ci, int T){
    long n = (long)Co * Ci * T;
    k_wtrans<<<ew(n), 256, 0, stream>>>(w, dst, Co, Ci, T, n);
  };
  wtr(F(27), wt_c1, 128, 256, 1);
  wtr(F(29), wt_c2, 128, 128, 9);
  wtr(F(31), wt_c3, 128, 128, 9);
  const int gw0[6] = {12, 14, 16, 18, 20, 22};          // layer gru  z1 r1 q1 z2 r2 q2
  const int gw1[6] = {33, 35, 37, 39, 41, 43};          // attup gru
  const float* gb0[6] = {F(13), F(15), F(17), F(19), F(21), F(23)};
  const float* gb1[6] = {F(34), F(36), F(38), F(40), F(42), F(44)};
  for (int i = 0; i < 6; ++i){ wtr(F(gw0[i]), wt_gru[i], 128, 256, 5);
                               wtr(F(gw1[i]), wt_gru[6 + i], 128, 256, 5); }

  // ---- input NCHW -> NHWC into att & value ----
  k_in_tr<<<ew(Bn * LC), 256, 0, stream>>>(F(0), att, val, C, L, Bn * LC);

  dim3 gN128(18, 2, Bn), gN512(18, 8, Bn), gNL(18, 36, Bn), gc(18, 2, Bn);

  for (int layer = 0; layer < 2; ++layer){
    // val_proj = value @ v_w^T ; transpose for the A@vp GEMM
    k_gemm<ACT_NONE><<<gN128, 256, 0, stream>>>(val, LC, F(3), 0, nullptr, vp, LC, L, C, C);
    k_transpose<<<ew(Bn * LC), 256, 0, stream>>>(vp, vpT, L, C, Bn * LC);
    // ap = relu(att@w1^T)@w2^T
    k_gemm<ACT_RELU><<<gN128, 256, 0, stream>>>(att, LC, F(1), 0, nullptr, t1, LC, L, C, C);
    k_gemm<ACT_NONE><<<gN128, 256, 0, stream>>>(t1, LC, F(2), 0, nullptr, ap, LC, L, C, C);
    // row stats + per-batch scale
    k_zero<<<1, 32, 0, stream>>>(sc, 16);
    k_rowstat<<<(Bn * L) / 8, 256, 0, stream>>>(ap, normfac, rowtmp, Bn * L);
    k_batchred<<<Bn, 256, 0, stream>>>(rowtmp, sc, L, 1, 1.0f / (float)(L * C));
    k_an<<<ew(Bn * LC), 256, 0, stream>>>(ap, normfac, an, Bn * LC);
    // S = an @ an^T  (L x L), A = exp(scale*S), global max, degree-normalize
    k_gemm<ACT_NONE><<<gNL, 256, 0, stream>>>(an, LC, an, LC, nullptr, Abig, LL, L, L, C);
    k_expmax<<<ew(Bn * LL), 256, 0, stream>>>(Abig, sc, (unsigned*)(sc + 2), Bn * LL, LL);
    k_rowsum<<<(Bn * L) / 8, 256, 0, stream>>>(Abig, rowtmp, L, Bn * L);
    k_dg<<<ew(Bn * L), 256, 0, stream>>>(rowtmp, (const unsigned*)(sc + 2), Dg, Bn * L);
    k_anorm<<<ew(Bn * LL), 256, 0, stream>>>(Abig, Dg, (const unsigned*)(sc + 2), Bn * LL, L, LL);
    // msg = LN( (A @ vp) @ merge^T )
    k_gemm<ACT_NONE><<<gN128, 256, 0, stream>>>(Abig, LL, vpT, LC, nullptr, bufA, LC, L, C, L);
    k_gemm<ACT_NONE><<<gN128, 256, 0, stream>>>(bufA, LC, F(4), 0, nullptr, bufB, LC, L, C, C);
    k_ln<<<(Bn * L) / 8, 256, 0, stream>>>(bufB, F(5), F(6), bufA, Bn * L);
    // MLP: cat(value,msg) -> 512 -> 512 -> 128 -> LN
    k_cat2<<<ew(Bn * L2C), 256, 0, stream>>>(val, bufA, cat0, (long)Bn * L, C);
    k_gemm<ACT_GELU><<<gN512, 256, 0, stream>>>(cat0, L2C, F(7), 0, nullptr, h1, L512, L, 512, 256);
    k_gemm<ACT_GELU><<<gN512, 256, 0, stream>>>(h1, L512, F(8), 0, nullptr, h2, L512, L, 512, 512);
    k_gemm<ACT_NONE><<<gN128, 256, 0, stream>>>(h2, L512, F(9), 0, nullptr, bufB, LC, L, C, 512);
    k_ln<<<(Bn * L) / 8, 256, 0, stream>>>(bufB, F(10), F(11), bufC, Bn * L);   // msg2
    // AttUp: proj -> LN -> cat(att,.) -> conv1(1x1,gelu) -> conv2/conv3(3x3,gelu)
    k_gemm<ACT_NONE><<<gN128, 256, 0, stream>>>(bufC, LC, F(24), 0, nullptr, bufD, LC, L, C, C);
    k_ln<<<(Bn * L) / 8, 256, 0, stream>>>(bufD, F(25), F(26), bufE, Bn * L);
    k_cat2<<<ew(Bn * L2C), 256, 0, stream>>>(att, bufE, cat1, (long)Bn * L, C);
    k_conv<1, ACT_GELU><<<gc, 256, 0, stream>>>(cat1, L2C, 256, wt_c1, F(28), zpage, bufD, LC, C, taps1, HH, WW);
    k_conv<9, ACT_GELU><<<gc, 256, 0, stream>>>(bufD, LC, 128, wt_c2, F(30), zpage, bufE, LC, C, taps3, HH, WW);
    k_conv<9, ACT_GELU><<<gc, 256, 0, stream>>>(bufE, LC, 128, wt_c3, F(32), zpage, bufD, LC, C, taps3, HH, WW);
    // att <- sepGRU(att, m); value <- sepGRU(value, msg2)
    run_sep_gru(stream, att, bufD, &wt_gru[6], gb1, zpage, cat0, cat1, zb, rb, qb, tapsH, tapsV);
    run_sep_gru(stream, val, bufC, &wt_gru[0], gb0, zpage, cat0, cat1, zb, rb, qb, tapsH, tapsV);
  }

  // ---- reprojection + global-mean normalization + NHWC->NCHW ----
  k_gemm<ACT_GELU><<<gN128, 256, 0, stream>>>(val, LC, F(45), 0, F(46), bufD, LC, L, C, C);
  k_gemm<ACT_NONE><<<gN128, 256, 0, stream>>>(bufD, LC, F(47), 0, F(48), bufE, LC, L, C, C);
  k_rowsum<<<(Bn * L) / 8, 256, 0, stream>>>(bufE, rowtmp, C, Bn * L);
  k_batchred<<<Bn, 256, 0, stream>>>(rowtmp, sc + 3, L, 0, 0.0f);
  k_final<<<ew(Bn * LC), 256, 0, stream>>>(bufE, sc + 3, F(49), F(50), (float*)d_out,
                                           C, L, Bn * LC, 1.0f / (float)(L * C));
}